// PLFNet_81063212745201
// MI455X (gfx1250) — compile-verified
//
#include <hip/hip_runtime.h>
#include <stdint.h>

// ---------------------------------------------------------------------------
// PLF (piecewise-linear function) evaluation, MI455X / gfx1250.
// out[i] = lerp(ctrl[i][left], ctrl[i][left+1], w) with
//   left = clip(floor(p/0.8 + 2.5), 0, 4),  w = (p+2)/0.8 - left
//
// Pure streaming kernel (~668 MB @ 23.3 TB/s ~= 29 us). Strategy:
//  - per-wave 128-element tiles; ctrl tile (3072 B) staged into LDS with
//    CDNA5 async global->LDS b128 copies (ASYNCcnt), double buffered,
//    th:TH_LOAD_NT so the single-touch 500 MB ctrl stream skips L2 residency
//  - 2-point gather from LDS (ds_load_2addr), cheap vs the AoS-24B layout
//  - non-temporal b128 loads/stores for the p / out streams
//  - bucketize via fmaf(p, 1.25f, 2.5f): 1.25f exact, <=1 ulp vs reference
//    divide; bucket flips only at knots where the PLF is continuous
// ---------------------------------------------------------------------------

typedef float v4f __attribute__((ext_vector_type(4)));

#define PLF_CTRLF  6                       // NUM_PCS + 1 floats per element
#define WAVE_TILE  128                     // elements per wave-tile
#define TILE_F     (WAVE_TILE * PLF_CTRLF) // 768 floats
#define TILE_B     (TILE_F * 4)            // 3072 bytes
#define WPB        8                       // waves per block (wave32)
#define BLOCK      256

// Copy one 3072-byte ctrl tile (128 elems x 24 B) global -> LDS.
// Each lane moves 96 contiguous bytes via six async b128 ops; IOFFSET
// advances BOTH the global and the LDS address, so layout is identity.
__device__ __forceinline__ void async_ctrl_tile(uint32_t lds_byte_addr,
                                                const float* gsrc, int lane) {
    uint64_t ga = (uint64_t)(uintptr_t)(gsrc + lane * 24);   // 96 B per lane
    uint32_t la = lds_byte_addr + (uint32_t)(lane * 96);
    asm volatile(
        "global_load_async_to_lds_b128 %0, %1, off th:TH_LOAD_NT\n\t"
        "global_load_async_to_lds_b128 %0, %1, off offset:16 th:TH_LOAD_NT\n\t"
        "global_load_async_to_lds_b128 %0, %1, off offset:32 th:TH_LOAD_NT\n\t"
        "global_load_async_to_lds_b128 %0, %1, off offset:48 th:TH_LOAD_NT\n\t"
        "global_load_async_to_lds_b128 %0, %1, off offset:64 th:TH_LOAD_NT\n\t"
        "global_load_async_to_lds_b128 %0, %1, off offset:80 th:TH_LOAD_NT"
        :
        : "v"(la), "v"(ga)
        : "memory");
}

__device__ __forceinline__ float plf_eval(float p, const float* c) {
    float t  = fmaf(p, 1.25f, 2.5f);       // == p/0.8 + 2.5 (<=1 ulp)
    float fl = floorf(t);
    fl = fminf(fmaxf(fl, 0.0f), 4.0f);     // clip to [0, NUM_PCS-1]
    int   left = (int)fl;
    float w  = t - fl;                     // == (p+2)/0.8 - left (<=1 ulp)
    float cl = c[left];
    float cr = c[left + 1];
    return cl + w * (cr - cl);
}

__global__ __launch_bounds__(BLOCK) void plf_kernel(const float* __restrict__ p,
                                                    const float* __restrict__ c,
                                                    float* __restrict__ out,
                                                    long long nTiles) {
    __shared__ float smem[WPB * 2 * TILE_F];   // 48 KB: double buffer per wave

    const int lane = threadIdx.x & 31;         // wave32
    const int wave = threadIdx.x >> 5;

    // Low 32 bits of a generic LDS pointer == LDS byte offset (AS3->generic
    // lowering is {aperture_hi, lds_offset}).
    const uint32_t ldsWave = (uint32_t)(size_t)smem + (uint32_t)(wave * 2 * TILE_B);
    const float*   swave   = smem + wave * 2 * TILE_F;

    long long       tile   = (long long)blockIdx.x * WPB + wave;
    const long long stride = (long long)gridDim.x * WPB;
    if (tile >= nTiles) return;

    int cur = 0;
    async_ctrl_tile(ldsWave, c + tile * TILE_F, lane);   // prologue: buf 0

    for (; tile < nTiles; tile += stride) {
        long long nxt = tile + stride;                   // wave-uniform branch
        if (nxt < nTiles) {
            // prefetch next tile into the other buffer, then wait for current
            async_ctrl_tile(ldsWave + (uint32_t)((cur ^ 1) * TILE_B),
                            c + nxt * TILE_F, lane);
            asm volatile("s_wait_asynccnt 6" ::: "memory");  // 6 newest remain
        } else {
            asm volatile("s_wait_asynccnt 0" ::: "memory");
        }

        // lane handles elements (4*lane .. 4*lane+3) of this 128-element tile
        const float* cb   = swave + cur * TILE_F + lane * (4 * PLF_CTRLF);
        long long    base = tile * WAVE_TILE + lane * 4;

        v4f pv = __builtin_nontemporal_load((const v4f*)(p + base));
        v4f ov;
        ov.x = plf_eval(pv.x, cb);                 // LDS gathers: ds_load_2addr
        ov.y = plf_eval(pv.y, cb + PLF_CTRLF);
        ov.z = plf_eval(pv.z, cb + 2 * PLF_CTRLF);
        ov.w = plf_eval(pv.w, cb + 3 * PLF_CTRLF);
        __builtin_nontemporal_store(ov, (v4f*)(out + base));

        cur ^= 1;
    }
}

// Scalar tail (n % 128 elements) — direct global loads, no staging.
__global__ __launch_bounds__(BLOCK) void plf_tail(const float* __restrict__ p,
                                                  const float* __restrict__ c,
                                                  float* __restrict__ out,
                                                  long long start, long long n) {
    long long i = start + (long long)blockIdx.x * blockDim.x + threadIdx.x;
    if (i >= n) return;
    out[i] = plf_eval(p[i], c + i * PLF_CTRLF);
}

extern "C" void kernel_launch(void* const* d_in, const int* in_sizes, int n_in,
                              void* d_out, int out_size, void* d_ws, size_t ws_size,
                              hipStream_t stream) {
    (void)n_in; (void)out_size; (void)d_ws; (void)ws_size;

    const float* param1 = (const float*)d_in[0];
    const float* param2 = (const float*)d_in[1];
    const float* ctrl1  = (const float*)d_in[2];
    const float* ctrl2  = (const float*)d_in[3];
    float*       out    = (float*)d_out;

    const long long n1 = in_sizes[0];   // 4096*4096
    const long long n2 = in_sizes[1];   // 1000*4096

    const long long t1 = n1 / WAVE_TILE, r1 = n1 % WAVE_TILE;
    const long long t2 = n2 / WAVE_TILE, r2 = n2 % WAVE_TILE;

    if (t1 > 0) {
        long long b = (t1 + WPB - 1) / WPB; if (b > 2048) b = 2048;
        plf_kernel<<<(int)b, BLOCK, 0, stream>>>(param1, ctrl1, out, t1);
    }
    if (r1 > 0) {
        plf_tail<<<1, BLOCK, 0, stream>>>(param1, ctrl1, out, t1 * WAVE_TILE, n1);
    }

    float* out2 = out + n1;
    if (t2 > 0) {
        long long b = (t2 + WPB - 1) / WPB; if (b > 2048) b = 2048;
        plf_kernel<<<(int)b, BLOCK, 0, stream>>>(param2, ctrl2, out2, t2);
    }
    if (r2 > 0) {
        plf_tail<<<1, BLOCK, 0, stream>>>(param2, ctrl2, out2, t2 * WAVE_TILE, n2);
    }
}